// SSMBlock_77996606095827
// MI455X (gfx1250) — compile-verified
//
#include <hip/hip_runtime.h>
#include <math.h>

typedef __attribute__((ext_vector_type(2))) float v2f;
typedef __attribute__((ext_vector_type(8))) float v8f;

#define D_MODEL 1024
#define D_STATE 16
#define D_CONV  4
#define D_INNER 2048
#define DT_RANK 64
#define B_SZ    4
#define L_SEQ   2048
#define NROWS   (B_SZ * L_SEQ)   // 8192

// ---------------- LayerNorm over D_MODEL ----------------
__global__ __launch_bounds__(256) void ln_kernel(const float* __restrict__ x,
                                                 const float* __restrict__ w,
                                                 const float* __restrict__ b,
                                                 float* __restrict__ out) {
    int row = blockIdx.x;
    int tid = threadIdx.x;
    const float* xr = x + (size_t)row * D_MODEL;
    float v[4];
    float s = 0.f;
#pragma unroll
    for (int i = 0; i < 4; ++i) { v[i] = xr[tid + i * 256]; s += v[i]; }
    __shared__ float red[256];
    red[tid] = s; __syncthreads();
    for (int off = 128; off > 0; off >>= 1) {
        if (tid < off) red[tid] += red[tid + off];
        __syncthreads();
    }
    float mu = red[0] * (1.f / D_MODEL);
    __syncthreads();
    float s2 = 0.f;
#pragma unroll
    for (int i = 0; i < 4; ++i) { float d = v[i] - mu; s2 += d * d; }
    red[tid] = s2; __syncthreads();
    for (int off = 128; off > 0; off >>= 1) {
        if (tid < off) red[tid] += red[tid + off];
        __syncthreads();
    }
    float inv = rsqrtf(red[0] * (1.f / D_MODEL) + 1e-5f);
    float* outr = out + (size_t)row * D_MODEL;
#pragma unroll
    for (int i = 0; i < 4; ++i) {
        int c = tid + i * 256;
        outr[c] = (v[i] - mu) * inv * w[c] + b[c];
    }
}

// ---------------- fp32 WMMA GEMM: C = A(MxK) * W(NxK)^T + bias ----------------
// Requirements: M % 128 == 0, K % 32 == 0 (true for all call sites); N arbitrary.
// mode: 0 = none, 1 = +bias, 2 = +bias then softplus
// LDS tiles stored as K-pair float2's so a WMMA fragment is a single ds_load_b64.
#define BM 128
#define BN 128
#define BK 32
#define LPAD 16  // float2 row stride 144 -> +1 pair-row shifts 32 banks: half-waves disjoint

__global__ __launch_bounds__(256) void gemm_wmma_f32(const float* __restrict__ A,
                                                     const float* __restrict__ W,
                                                     const float* __restrict__ bias,
                                                     float* __restrict__ C,
                                                     int M, int N, int K, int mode) {
    __shared__ v2f As2[2][BK / 2][BM + LPAD];
    __shared__ v2f Bs2[2][BK / 2][BN + LPAD];
    int tid  = threadIdx.x;
    int lane = tid & 31;
    int wave = tid >> 5;
    int wmRow = wave & 3;   // 4 M-waves, 32 rows each
    int wnCol = wave >> 2;  // 2 N-waves, 64 cols each
    int half = lane >> 4;   // K half-wave select: pair (k,k+1) vs (k+2,k+3)
    int l15  = lane & 15;
    int mBase = blockIdx.y * BM;
    int nBase = blockIdx.x * BN;

    // staging registers: 4x float4 per tile per thread
    float4 ar[4], br[4];
    // staging index map: lin = tid + i*256 in [0,1024); row = lin>>3; kq = (lin&7)*4
    int sRow = tid >> 3;          // base row (+32 per i)
    int sKq  = (tid & 7) * 4;     // float column within BK (multiple of 4)
    int sKp  = sKq >> 1;          // K-pair index (multiple of 2)

    v8f acc[2][4] = {};

    int nstages = K / BK;

    // prologue: stage 0 global loads
#pragma unroll
    for (int i = 0; i < 4; ++i) {
        int mm = sRow + i * 32;
        ar[i] = *(const float4*)(A + (size_t)(mBase + mm) * K + sKq);
        int gn = nBase + sRow + i * 32;
        if (gn < N) br[i] = *(const float4*)(W + (size_t)gn * K + sKq);
        else        br[i] = make_float4(0.f, 0.f, 0.f, 0.f);
    }
#pragma unroll
    for (int i = 0; i < 4; ++i) {
        int mm = sRow + i * 32;
        v2f p0, p1, q0, q1;
        p0.x = ar[i].x; p0.y = ar[i].y; p1.x = ar[i].z; p1.y = ar[i].w;
        q0.x = br[i].x; q0.y = br[i].y; q1.x = br[i].z; q1.y = br[i].w;
        As2[0][sKp][mm] = p0; As2[0][sKp + 1][mm] = p1;
        Bs2[0][sKp][mm] = q0; Bs2[0][sKp + 1][mm] = q1;
    }
    __syncthreads();

    for (int s = 0; s < nstages; ++s) {
        int cur = s & 1;
        // issue global loads for next stage (overlap with WMMA compute)
        if (s + 1 < nstages) {
            int k0 = (s + 1) * BK;
#pragma unroll
            for (int i = 0; i < 4; ++i) {
                int mm = sRow + i * 32;
                ar[i] = *(const float4*)(A + (size_t)(mBase + mm) * K + k0 + sKq);
                int gn = nBase + sRow + i * 32;
                if (gn < N) br[i] = *(const float4*)(W + (size_t)gn * K + k0 + sKq);
                else        br[i] = make_float4(0.f, 0.f, 0.f, 0.f);
            }
        }

        // compute on current stage: 8 k-steps x 8 WMMA, fragments are single b64 loads
#pragma unroll
        for (int kk = 0; kk < BK; kk += 4) {
            int kp = (kk >> 1) + half;
            v2f a[2];
#pragma unroll
            for (int u = 0; u < 2; ++u)
                a[u] = As2[cur][kp][wmRow * 32 + u * 16 + l15];
#pragma unroll
            for (int t = 0; t < 4; ++t) {
                v2f b = Bs2[cur][kp][wnCol * 64 + t * 16 + l15];
#pragma unroll
                for (int u = 0; u < 2; ++u) {
                    acc[u][t] = __builtin_amdgcn_wmma_f32_16x16x4_f32(
                        false, a[u], false, b, (short)0, acc[u][t], false, false);
                }
            }
        }

        // store next stage into the other buffer
        if (s + 1 < nstages) {
            int nxt = cur ^ 1;
#pragma unroll
            for (int i = 0; i < 4; ++i) {
                int mm = sRow + i * 32;
                v2f p0, p1, q0, q1;
                p0.x = ar[i].x; p0.y = ar[i].y; p1.x = ar[i].z; p1.y = ar[i].w;
                q0.x = br[i].x; q0.y = br[i].y; q1.x = br[i].z; q1.y = br[i].w;
                As2[nxt][sKp][mm] = p0; As2[nxt][sKp + 1][mm] = p1;
                Bs2[nxt][sKp][mm] = q0; Bs2[nxt][sKp + 1][mm] = q1;
            }
        }
        __syncthreads();
    }

    // Epilogue per ISA C/D layout: VGPR r -> row r + 8*half
#pragma unroll
    for (int u = 0; u < 2; ++u) {
#pragma unroll
        for (int t = 0; t < 4; ++t) {
            int col = nBase + wnCol * 64 + t * 16 + l15;
            if (col >= N) continue;
            float bv = (bias != nullptr) ? bias[col] : 0.f;
#pragma unroll
            for (int r = 0; r < 8; ++r) {
                int row = mBase + wmRow * 32 + u * 16 + half * 8 + r;
                float v = acc[u][t][r] + bv;
                if (mode == 2) v = (v > 20.f) ? v : log1pf(expf(v));
                C[(size_t)row * N + col] = v;
            }
        }
    }
}

// ---------------- fused causal depthwise conv (K=4) + SiLU for both halves ----------------
__global__ __launch_bounds__(256) void conv_silu_kernel(const float* __restrict__ xz,
                                                        const float* __restrict__ cw,
                                                        const float* __restrict__ cb,
                                                        const float* __restrict__ gw,
                                                        const float* __restrict__ gb,
                                                        float* __restrict__ xs,
                                                        float* __restrict__ zal) {
    int idx = blockIdx.x * blockDim.x + threadIdx.x;
    int total = B_SZ * L_SEQ * D_INNER;
    if (idx >= total) return;
    int i = idx % D_INNER;
    int l = (idx / D_INNER) % L_SEQ;
    int b = idx / (D_INNER * L_SEQ);
    float ax = cb[i], az = gb[i];
#pragma unroll
    for (int k = 0; k < D_CONV; ++k) {
        int t = l - (D_CONV - 1) + k;
        if (t < 0) continue;
        size_t base = (size_t)(b * L_SEQ + t) * (2 * D_INNER);
        ax = fmaf(cw[i * D_CONV + k], xz[base + i], ax);
        az = fmaf(gw[i * D_CONV + k], xz[base + D_INNER + i], az);
    }
    xs[idx]  = ax / (1.f + expf(-ax));
    zal[idx] = az / (1.f + expf(-az));
}

// ---------------- dt mean over (batch*L) ----------------
__global__ __launch_bounds__(256) void dt_mean_kernel(const float* __restrict__ dt,
                                                      float* __restrict__ dt_avg) {
    int ch = blockIdx.x * blockDim.x + threadIdx.x;
    if (ch >= D_INNER) return;
    float s = 0.f;
    for (int r = 0; r < NROWS; ++r) s += dt[(size_t)r * D_INNER + ch];
    dt_avg[ch] = s * (1.f / NROWS);
}

// ---------------- discretization: A_bar, B_bar ----------------
__global__ __launch_bounds__(256) void discretize_kernel(const float* __restrict__ A_log,
                                                         const float* __restrict__ Bm,
                                                         const float* __restrict__ dt_avg,
                                                         float* __restrict__ Abar,
                                                         float* __restrict__ Bbar) {
    int idx = blockIdx.x * blockDim.x + threadIdx.x;
    if (idx >= D_INNER * D_STATE) return;
    int i = idx / D_STATE;
    float A   = -expf(A_log[idx]);
    float dtv = dt_avg[i];
    float Ae  = A * dtv;
    float Ab  = expf(Ae);
    bool small = fabsf(Ae) < 1e-6f;
    float expterm = small ? Ae : (Ab - 1.f);
    float Ainv    = 1.f / (A + 1e-8f);
    float safe    = small ? dtv : Ainv * expterm;
    Abar[idx] = Ab;
    Bbar[idx] = safe * Bm[idx];
}

// ---------------- time-invariant SSM scan, fused gate/D epilogue ----------------
__global__ __launch_bounds__(256) void scan_kernel(const float* __restrict__ xs,
                                                   const float* __restrict__ zal,
                                                   const float* __restrict__ Abar,
                                                   const float* __restrict__ Bbar,
                                                   const float* __restrict__ Cm,
                                                   const float* __restrict__ Dv,
                                                   float* __restrict__ G) {
    int gid = blockIdx.x * blockDim.x + threadIdx.x;
    if (gid >= B_SZ * D_INNER) return;
    int i = gid % D_INNER;
    int b = gid / D_INNER;
    float Ab[D_STATE], Bb[D_STATE], Cv[D_STATE], h[D_STATE];
#pragma unroll
    for (int d = 0; d < D_STATE; ++d) {
        Ab[d] = Abar[i * D_STATE + d];
        Bb[d] = Bbar[i * D_STATE + d];
        Cv[d] = Cm[i * D_STATE + d];
        h[d]  = 0.f;
    }
    float dfac = 1.f + Dv[i];
    size_t base = (size_t)b * L_SEQ * D_INNER + i;
    for (int t = 0; t < L_SEQ; ++t) {
        size_t o = base + (size_t)t * D_INNER;
        float xv = xs[o];
        float y = 0.f;
#pragma unroll
        for (int d = 0; d < D_STATE; ++d) {
            h[d] = fmaf(Ab[d], h[d], Bb[d] * xv);
            y = fmaf(h[d], Cv[d], y);
        }
        G[o] = y * zal[o] * dfac;
    }
}

extern "C" void kernel_launch(void* const* d_in, const int* in_sizes, int n_in,
                              void* d_out, int out_size, void* d_ws, size_t ws_size,
                              hipStream_t stream) {
    const float* x          = (const float*)d_in[0];
    const float* ln_w       = (const float*)d_in[1];
    const float* ln_b       = (const float*)d_in[2];
    const float* in_proj_w  = (const float*)d_in[3];
    const float* in_proj_b  = (const float*)d_in[4];
    const float* conv_w     = (const float*)d_in[5];
    const float* conv_b     = (const float*)d_in[6];
    const float* gate_w     = (const float*)d_in[7];
    const float* gate_b     = (const float*)d_in[8];
    const float* A_log      = (const float*)d_in[9];
    const float* B_mat      = (const float*)d_in[10];
    const float* C_mat      = (const float*)d_in[11];
    const float* D_vec      = (const float*)d_in[12];
    const float* A_proj_w   = (const float*)d_in[13];
    const float* dt_proj_w  = (const float*)d_in[14];
    const float* dt_proj_b  = (const float*)d_in[15];
    const float* out_proj_w = (const float*)d_in[16];
    const float* out_proj_b = (const float*)d_in[17];

    // Workspace layout (floats), with dead-buffer reuse; peak 72M floats = 288MB.
    float* ws = (float*)d_ws;
    const size_t M1 = 1024 * 1024;
    float* xn  = ws;            // [0, 8M)   : LN output (dead after in_proj)
    float* xz  = ws + 8 * M1;   // [8M, 40M) : in_proj output (dead after conv)
    float* xs  = ws + 40 * M1;  // [40M, 56M)
    float* zal = ws + 56 * M1;  // [56M, 72M)
    // reuse of dead regions:
    float* A_in   = ws;                          // [0, 0.5M)   (in xn region)
    float* dt_avg = ws + 4 * M1;                 // 2048        (in xn region)
    float* Abar   = ws + 4 * M1 + 4096;          // 32768       (in xn region)
    float* Bbar   = ws + 4 * M1 + 4096 + 32768;  // 32768       (in xn region)
    float* dtb    = ws + 8 * M1;                 // [8M, 24M)   (in xz region)
    float* G      = ws + 24 * M1;                // [24M, 40M)  (in xz region)

    // 1) LayerNorm
    ln_kernel<<<NROWS, 256, 0, stream>>>(x, ln_w, ln_b, xn);

    // 2) in_proj: (8192 x 1024) x (1024 x 4096) + bias
    {
        dim3 g((2 * D_INNER + BN - 1) / BN, NROWS / BM);
        gemm_wmma_f32<<<g, 256, 0, stream>>>(xn, in_proj_w, in_proj_b, xz,
                                             NROWS, 2 * D_INNER, D_MODEL, 1);
    }

    // 3) causal depthwise conv + SiLU on both halves
    {
        int total = B_SZ * L_SEQ * D_INNER;
        conv_silu_kernel<<<(total + 255) / 256, 256, 0, stream>>>(
            xz, conv_w, conv_b, gate_w, gate_b, xs, zal);
    }

    // 4) A_proj: (8192 x 2048) x (2048 x 64), no bias
    {
        dim3 g((DT_RANK + BN - 1) / BN, NROWS / BM);
        gemm_wmma_f32<<<g, 256, 0, stream>>>(xs, A_proj_w, nullptr, A_in,
                                             NROWS, DT_RANK, D_INNER, 0);
    }

    // 5) dt_proj + softplus: (8192 x 64) x (64 x 2048) + bias
    {
        dim3 g((D_INNER + BN - 1) / BN, NROWS / BM);
        gemm_wmma_f32<<<g, 256, 0, stream>>>(A_in, dt_proj_w, dt_proj_b, dtb,
                                             NROWS, D_INNER, DT_RANK, 2);
    }

    // 6) dt mean over batch*time
    dt_mean_kernel<<<(D_INNER + 255) / 256, 256, 0, stream>>>(dtb, dt_avg);

    // 7) discretization
    discretize_kernel<<<(D_INNER * D_STATE + 255) / 256, 256, 0, stream>>>(
        A_log, B_mat, dt_avg, Abar, Bbar);

    // 8) recurrence + gate + D epilogue -> G
    scan_kernel<<<(B_SZ * D_INNER + 255) / 256, 256, 0, stream>>>(
        xs, zal, Abar, Bbar, C_mat, D_vec, G);

    // 9) out_proj: (8192 x 2048) x (2048 x 1024) + bias -> d_out
    {
        dim3 g((D_MODEL + BN - 1) / BN, NROWS / BM);
        gemm_wmma_f32<<<g, 256, 0, stream>>>(G, out_proj_w, out_proj_b,
                                             (float*)d_out, NROWS, D_MODEL, D_INNER, 1);
    }
}